// Model_35502199669508
// MI455X (gfx1250) — compile-verified
//
#include <hip/hip_runtime.h>
#include <stdint.h>

#define B_ 32
#define S_ 256
#define T_ 2048
#define M_ 80
#define NEGF (-3.4028234663852886e38f)

typedef __attribute__((ext_vector_type(2))) float v2f;
typedef __attribute__((ext_vector_type(8))) float v8f;

// ---------------------------------------------------------------------------
// K1: masked cumsum of durations per batch (Hillis-Steele in LDS)
// ---------------------------------------------------------------------------
__global__ void k_scan(const int* __restrict__ dur, const int* __restrict__ textlen,
                       int* __restrict__ cums) {
    int b = blockIdx.x, s = threadIdx.x;
    __shared__ int sm[S_];
    int tl = textlen[b];
    sm[s] = (s < tl) ? dur[b * S_ + s] : 0;
    __syncthreads();
    for (int off = 1; off < S_; off <<= 1) {
        int add = (s >= off) ? sm[s - off] : 0;
        __syncthreads();
        sm[s] += add;
        __syncthreads();
    }
    cums[b * S_ + s] = sm[s];
}

// ---------------------------------------------------------------------------
// K2: attn_align[b,t,s] = (cums[s-1] <= t < cums[s]); align_mask[b,t] = t < cums[S-1]
// One block per (b, 64-row t tile); thread = s; 1KB coalesced row writes.
// ---------------------------------------------------------------------------
__global__ void k_align(const int* __restrict__ cums, float* __restrict__ attn_align,
                        float* __restrict__ align_mask) {
    int b = blockIdx.x;
    int t0 = blockIdx.y * 64;
    int s = threadIdx.x;
    int chi = cums[b * S_ + s];
    int clo = (s == 0) ? 0 : cums[b * S_ + s - 1];
    int ctot = cums[b * S_ + S_ - 1];
    float* ob = attn_align + ((size_t)b * T_ + t0) * S_ + s;
    for (int dt = 0; dt < 64; ++dt) {
        int t = t0 + dt;
        ob[(size_t)dt * S_] = (t >= clo && t < chi) ? 1.f : 0.f;
        if (s == 0) align_mask[(size_t)b * T_ + t] = (t < ctot) ? 1.f : 0.f;
    }
}

// ---------------------------------------------------------------------------
// K3: MAS forward DP. One wave32 per batch; lane owns 8 contiguous columns.
// Serial chain per step: one __shfl_up + a few VALU. ll row j+1 prefetched.
// Direction bits: 1 byte per lane per step -> dirs[b][j][32].
// ---------------------------------------------------------------------------
__global__ void __launch_bounds__(32) k_mas_fwd(const float* __restrict__ ll,
                                                const int* __restrict__ textlen,
                                                const int* __restrict__ mellen,
                                                unsigned char* __restrict__ dirs) {
    int b = blockIdx.x;
    int lane = threadIdx.x;
    int tl = textlen[b], ml = mellen[b];
    int sbase = lane * 8;

    float prob[8];
#pragma unroll
    for (int k = 0; k < 8; ++k) prob[k] = 0.f;

    const float* llb = ll + (size_t)b * T_ * S_ + sbase;
    float4 c0 = *(const float4*)(llb);
    float4 c1 = *(const float4*)(llb + 4);
    unsigned char* db = dirs + (size_t)b * T_ * 32 + lane;

    for (int j = 0; j < T_; ++j) {
        int jn = (j + 1 < T_) ? j + 1 : j;
        float4 n0 = *(const float4*)(llb + (size_t)jn * S_);
        float4 n1 = *(const float4*)(llb + (size_t)jn * S_ + 4);

        float left = __shfl_up(prob[7], 1, 32);
        if (lane == 0) left = NEGF;

        float lv[8] = {c0.x, c0.y, c0.z, c0.w, c1.x, c1.y, c1.z, c1.w};
        bool rowin = (j < ml);
        unsigned byteval = 0;
        float prevv = left;
#pragma unroll
        for (int k = 0; k < 8; ++k) {
            float p = prob[k];
            bool m = (p >= prevv);                 // max_mask (ties keep prob)
            byteval |= ((unsigned)m) << k;
            float pm = m ? p : prevv;
            int s = sbase + k;
            float lm = (rowin && (s < tl)) ? lv[k] : 0.f;  // ll * attnmask
            prob[k] = (s <= j) ? (pm + lm) : NEGF;
            prevv = p;                             // old prob[k] feeds column k+1
        }
        db[(size_t)j * 32] = (unsigned char)byteval;
        c0 = n0;
        c1 = n1;
    }
}

// ---------------------------------------------------------------------------
// K4: backtrack; 32 independent chains in one wave (lane = batch).
// Direction row (32B, address depends only on j) is prefetched -> the
// index-dependent bit-select is pure VALU, keeping loads off the serial chain.
// ---------------------------------------------------------------------------
__global__ void __launch_bounds__(32) k_mas_bwd(const unsigned char* __restrict__ dirs,
                                                const int* __restrict__ textlen,
                                                const int* __restrict__ mellen,
                                                int* __restrict__ idxout) {
    int b = threadIdx.x;
    int tl = textlen[b], ml = mellen[b];
    int idx = tl - 1;
    const uint4* dr = (const uint4*)(dirs + (size_t)b * T_ * 32);
    uint4 a0 = dr[2 * (T_ - 1)];
    uint4 a1 = dr[2 * (T_ - 1) + 1];

    for (int j = T_ - 1; j >= 0; --j) {
        int jn = (j > 0) ? (j - 1) : 0;
        uint4 p0 = dr[2 * jn];
        uint4 p1 = dr[2 * jn + 1];

        idxout[b * T_ + j] = idx;  // one_hot row recorded before index update

        int sc = idx < 0 ? 0 : (idx > (S_ - 1) ? (S_ - 1) : idx);  // JAX clip
        int wi = sc >> 5;
        unsigned w = (wi & 4) ? ((wi & 2) ? ((wi & 1) ? a1.w : a1.z) : ((wi & 1) ? a1.y : a1.x))
                              : ((wi & 2) ? ((wi & 1) ? a0.w : a0.z) : ((wi & 1) ? a0.y : a0.x));
        int dbit = (int)((w >> (sc & 31)) & 1u);
        int d = ((j < ml) && (sc < tl)) ? dbit : 1;  // outside mask: stay
        idx += d - 1;
        a0 = p0;
        a1 = p1;
    }
}

// ---------------------------------------------------------------------------
// K5: attn_mas[b,t,s] = (s==idx[b,t]) * attnmask; coalesced 1KB row writes.
// ---------------------------------------------------------------------------
__global__ void k_mas_write(const int* __restrict__ idxout, const int* __restrict__ textlen,
                            const int* __restrict__ mellen, float* __restrict__ attn_mas) {
    int b = blockIdx.x;
    int t0 = blockIdx.y * 8;
    int s = threadIdx.x;
    int tl = textlen[b], ml = mellen[b];
    bool sin_ = (s < tl);
    for (int dt = 0; dt < 8; ++dt) {
        int t = t0 + dt;
        int idx = idxout[b * T_ + t];
        attn_mas[((size_t)b * T_ + t) * S_ + s] = (sin_ && (t < ml) && (s == idx)) ? 1.f : 0.f;
    }
}

// ---------------------------------------------------------------------------
// K6: expanded = attn_mas @ mean via v_wmma_f32_16x16x4_f32 (exact in f32:
// one-hot A rows synthesized in registers from idxout; no attn_mas re-read).
// One wave per 16(t) x 16(m) output tile; 64 k-chunks of 4 over S=256.
// ISA layouts (7.12.2): A 16x4: lanes0-15 -> K=0,1 in v0,v1; lanes16-31 -> K=2,3.
//                      B 4x16:  row = vgpr + 2*half, col = lane&15.
//                      D 16x16: row = vgpr + 8*half, col = lane&15.
// ---------------------------------------------------------------------------
__global__ void __launch_bounds__(32) k_expand(const int* __restrict__ idxout,
                                               const float* __restrict__ mean,
                                               const int* __restrict__ textlen,
                                               const int* __restrict__ mellen,
                                               float* __restrict__ out) {
    int b = blockIdx.x;
    int trow = blockIdx.y * 16;
    int m0 = blockIdx.z * 16;
    int L = threadIdx.x;
    int half = L >> 4;
    int lr = L & 15;
    int tl = textlen[b], ml = mellen[b];

    int t = trow + lr;
    int iv = idxout[b * T_ + t];
    int tgt = ((t < ml) && (iv >= 0) && (iv < tl)) ? iv : -1;  // masked rows -> all-zero A row

    const float* mb = mean + (size_t)b * S_ * M_ + m0 + lr;
    v8f acc = {0.f, 0.f, 0.f, 0.f, 0.f, 0.f, 0.f, 0.f};

    for (int s0 = 0; s0 < S_; s0 += 4) {
        int k0 = s0 + 2 * half;
        v2f a, bmat;
        a.x = (tgt == k0) ? 1.f : 0.f;
        a.y = (tgt == k0 + 1) ? 1.f : 0.f;
        bmat.x = mb[(size_t)k0 * M_];
        bmat.y = mb[(size_t)(k0 + 1) * M_];
        acc = __builtin_amdgcn_wmma_f32_16x16x4_f32(false, a, false, bmat,
                                                    (short)0, acc, false, false);
    }

    float* ob = out + ((size_t)b * T_ + trow) * M_ + m0 + lr;
#pragma unroll
    for (int v = 0; v < 8; ++v) {
        int row = v + half * 8;
        ob[(size_t)row * M_] = acc[v];
    }
}

// ---------------------------------------------------------------------------
extern "C" void kernel_launch(void* const* d_in, const int* in_sizes, int n_in,
                              void* d_out, int out_size, void* d_ws, size_t ws_size,
                              hipStream_t stream) {
    const int* duration = (const int*)d_in[0];
    const float* ll = (const float*)d_in[1];
    const float* mean = (const float*)d_in[2];
    const int* textlen = (const int*)d_in[3];
    const int* mellen = (const int*)d_in[4];

    float* out = (float*)d_out;
    float* attn_align = out;                                     // [B,T,S]
    float* align_mask = attn_align + (size_t)B_ * T_ * S_;       // [B,T]
    float* attn_mas = align_mask + (size_t)B_ * T_;              // [B,T,S]
    float* expanded = attn_mas + (size_t)B_ * T_ * S_;           // [B,T,M]

    char* ws = (char*)d_ws;
    int* cums = (int*)ws;                                        // 32 KB
    unsigned char* dirs = (unsigned char*)(ws + 64 * 1024);      // B*T*32 = 2 MB
    int* idxout = (int*)(ws + 64 * 1024 + (size_t)B_ * T_ * 32); // 256 KB

    k_scan<<<B_, S_, 0, stream>>>(duration, textlen, cums);
    k_align<<<dim3(B_, T_ / 64), S_, 0, stream>>>(cums, attn_align, align_mask);
    k_mas_fwd<<<B_, 32, 0, stream>>>(ll, textlen, mellen, dirs);
    k_mas_bwd<<<1, 32, 0, stream>>>(dirs, textlen, mellen, idxout);
    k_mas_write<<<dim3(B_, T_ / 8), S_, 0, stream>>>(idxout, textlen, mellen, attn_mas);
    k_expand<<<dim3(B_, T_ / 16, M_ / 16), 32, 0, stream>>>(idxout, mean, textlen, mellen, expanded);
}